// LossYOLOv1_87325275062281
// MI455X (gfx1250) — compile-verified
//
#include <hip/hip_runtime.h>
#include <cstdint>

// ---------------- problem constants ----------------
#define G_      14
#define DIMS    90                       // 2*5 + 80
#define CELLS   (1024 * G_ * G_)         // 200704
#define TILE    32                       // cells per wave-tile
#define NTILES  (CELLS / TILE)           // 6272 (exact)
#define WPB     2                        // waves per block
#define TILE_BYTES (TILE * DIMS * 4)     // 11520 bytes per tensor per tile

typedef float v2f __attribute__((ext_vector_type(2)));
typedef float v8f __attribute__((ext_vector_type(8)));

__device__ __forceinline__ float sigm(float x) { return 1.0f / (1.0f + expf(-x)); }

__device__ __forceinline__ float pair_loss(const float* pb_, const float* gb_) {
    float dx = pb_[0] - gb_[0];
    float dy = pb_[1] - gb_[1];
    float dw = sqrtf(pb_[2]) - sqrtf(gb_[2]);
    float dh = sqrtf(pb_[3]) - sqrtf(gb_[3]);
    return (dx * dx + dy * dy) + (dw * dw + dh * dh);
}

__global__ __launch_bounds__(32 * WPB) void yolo_loss_partial(
    const float* __restrict__ P, const float* __restrict__ Gt, float* __restrict__ part)
{
    __shared__ float sh[WPB][2][TILE * DIMS];   // [wave][p/g][2880 floats] = 46080 B/block
    const int lane = threadIdx.x & 31;
    const int w    = threadIdx.x >> 5;
    const int tile = blockIdx.x * WPB + w;      // grid sized exactly: tile < NTILES

    // ---- stage one 32-cell tile of p and g into LDS (coalesced, async) ----
    {
        const uint64_t gp = (uint64_t)P  + (uint64_t)tile * TILE_BYTES;
        const uint64_t gg = (uint64_t)Gt + (uint64_t)tile * TILE_BYTES;
        const unsigned lp = (unsigned)(uintptr_t)&sh[w][0][0];
        const unsigned lg = (unsigned)(uintptr_t)&sh[w][1][0];
#if defined(__gfx1250__)
        // 11520 B = 22 * (32 lanes * 16 B) + 1 * (32 lanes * 8 B)
        #pragma unroll
        for (int i = 0; i < 22; ++i) {
            const unsigned o = (unsigned)i * 512u + (unsigned)lane * 16u;
            asm volatile("global_load_async_to_lds_b128 %0, %1, off"
                         :: "v"(lp + o), "v"(gp + o) : "memory");
            asm volatile("global_load_async_to_lds_b128 %0, %1, off"
                         :: "v"(lg + o), "v"(gg + o) : "memory");
        }
        {
            const unsigned o = 22u * 512u + (unsigned)lane * 8u;
            asm volatile("global_load_async_to_lds_b64 %0, %1, off"
                         :: "v"(lp + o), "v"(gp + o) : "memory");
            asm volatile("global_load_async_to_lds_b64 %0, %1, off"
                         :: "v"(lg + o), "v"(gg + o) : "memory");
        }
        asm volatile("s_wait_asynccnt 0" ::: "memory");
#else
        for (int i = lane; i < TILE * DIMS; i += 32) {
            sh[w][0][i] = ((const float*)gp)[i];
            sh[w][1][i] = ((const float*)gg)[i];
        }
#endif
    }

#define SP(i) sh[w][0][(i)]
#define SG(i) sh[w][1][(i)]

    // ---- per-cell scalar part: box + conf losses (lane == cell in tile) ----
    const int   c    = tile * TILE + lane;
    const int   rem  = c % (G_ * G_);
    const float colf = (float)(rem % G_);
    const float rowf = (float)(rem / G_);
    const int   base = lane * DIMS;

    const bool  mp = SG(base + 8) > 0.0f;
    const float m  = mp ? 1.0f : 0.0f;

    float po[2][4], go[2][4];
    #pragma unroll
    for (int b = 0; b < 2; ++b) {
        po[b][0] = sigm(SP(base + 4 * b + 0));
        po[b][1] = sigm(SP(base + 4 * b + 1));
        po[b][2] = SP(base + 4 * b + 2);
        po[b][3] = SP(base + 4 * b + 3);
        go[b][0] = SG(base + 4 * b + 0);
        go[b][1] = SG(base + 4 * b + 1);
        go[b][2] = SG(base + 4 * b + 2);
        go[b][3] = SG(base + 4 * b + 3);
    }

    float pl[2], pt[2], pr[2], pb[2], gl[2], gt[2], gr[2], gb[2], ap[2], ag[2];
    #pragma unroll
    for (int b = 0; b < 2; ++b) {
        float x = (po[b][0] + colf) / 14.0f, y = (po[b][1] + rowf) / 14.0f;
        pl[b] = x - po[b][2] * 0.5f; pr[b] = x + po[b][2] * 0.5f;
        pt[b] = y - po[b][3] * 0.5f; pb[b] = y + po[b][3] * 0.5f;
        ap[b] = (pr[b] - pl[b]) * (pb[b] - pt[b]);
        float gx = (go[b][0] + colf) / 14.0f, gy = (go[b][1] + rowf) / 14.0f;
        gl[b] = gx - go[b][2] * 0.5f; gr[b] = gx + go[b][2] * 0.5f;
        gt[b] = gy - go[b][3] * 0.5f; gb[b] = gy + go[b][3] * 0.5f;
        ag[b] = (gr[b] - gl[b]) * (gb[b] - gt[b]);
    }

    float iou[2][2];
    #pragma unroll
    for (int pi = 0; pi < 2; ++pi)
        #pragma unroll
        for (int gj = 0; gj < 2; ++gj) {
            float ltx = fmaxf(pl[pi], gl[gj]), lty = fmaxf(pt[pi], gt[gj]);
            float rbx = fminf(pr[pi], gr[gj]), rby = fminf(pb[pi], gb[gj]);
            float wi = fmaxf(rbx - ltx, 0.0f), hi = fmaxf(rby - lty, 0.0f);
            float inter = wi * hi;
            iou[pi][gj] = inter / (ap[pi] + ag[gj] - inter + 1e-12f);
        }
    // jnp.argmax over the p-box axis: first index wins ties
    const int ind0 = (iou[1][0] > iou[0][0]) ? 1 : 0;
    const int ind1 = (iou[1][1] > iou[0][1]) ? 1 : 0;

    float ps[2][4], gs[2][4];
    #pragma unroll
    for (int b = 0; b < 2; ++b) {
        ps[b][0] = po[b][0]; ps[b][1] = po[b][1];
        ps[b][2] = mp ? fabsf(po[b][2]) : 1.0f;
        ps[b][3] = mp ? fabsf(po[b][3]) : 1.0f;
        gs[b][0] = go[b][0]; gs[b][1] = go[b][1];
        gs[b][2] = mp ? go[b][2] : 1.0f;
        gs[b][3] = mp ? go[b][3] : 1.0f;
    }

    const float Lm0 = pair_loss(ps[ind0], gs[0]);
    const float Lm1 = pair_loss(ps[ind1], gs[1]);
    const float Li0 = pair_loss(ps[0], gs[0]);
    const float Li1 = pair_loss(ps[1], gs[1]);

    const bool condA = (go[0][0] == go[1][0]) && (go[0][1] == go[1][1]) &&
                       (go[0][2] == go[1][2]) && (go[0][3] == go[1][3]);
    const bool condB = (!condA) && (ind0 == ind1);

    const float lbox = condA ? Lm0 : (condB ? (Li0 + Li1) : (Lm0 + Lm1));

    const float cp0 = SP(base + 8), cp1 = SP(base + 9);
    const float pc1 = ind1 ? cp1 : cp0;
    const float csingle = (pc1 - 1.0f) * (pc1 - 1.0f);
    const float cboth   = (cp0 - 1.0f) * (cp0 - 1.0f) + (cp1 - 1.0f) * (cp1 - 1.0f);
    const float lcp = (condA || condB) ? csingle : cboth;
    const float lcn = cp0 * cp0 + cp1 * cp1;

    float lanePartial = 5.0f * (lbox * m) + lcp * m + 0.5f * (lcn * (1.0f - m));

    // ---- class loss via f32 WMMA: rowsum of squared masked diffs ----
#if defined(__gfx1250__) && __has_builtin(__builtin_amdgcn_wmma_f32_16x16x4_f32)
    {
        v8f acc0 = {}; v8f acc1 = {};
        const v2f bones = {1.0f, 1.0f};          // B all-ones -> D[i][j] = rowsum_i
        const int mrow = lane & 15;              // A: M = lane&15
        const int koff = (lane >> 4) << 1;       // A: K = {0,1} low half, {2,3} high half
        const float mA = (SG((0  + mrow) * DIMS + 8) > 0.0f) ? 1.0f : 0.0f;
        const float mB = (SG((16 + mrow) * DIMS + 8) > 0.0f) ? 1.0f : 0.0f;
        #pragma unroll
        for (int t = 0; t < 20; ++t) {           // 80 class dims / K=4
            const int d = 10 + 4 * t + koff;
            int r0 = (0 + mrow) * DIMS + d;
            float e0 = (SP(r0)     - SG(r0))     * mA;
            float e1 = (SP(r0 + 1) - SG(r0 + 1)) * mA;
            v2f a0 = {e0 * e0, e1 * e1};
            acc0 = __builtin_amdgcn_wmma_f32_16x16x4_f32(
                       false, a0, false, bones, (short)0, acc0, false, false);
            int r1 = (16 + mrow) * DIMS + d;
            float f0 = (SP(r1)     - SG(r1))     * mB;
            float f1 = (SP(r1 + 1) - SG(r1 + 1)) * mB;
            v2f a1 = {f0 * f0, f1 * f1};
            acc1 = __builtin_amdgcn_wmma_f32_16x16x4_f32(
                       false, a1, false, bones, (short)0, acc1, false, false);
        }
        float s = 0.0f;
        #pragma unroll
        for (int r = 0; r < 8; ++r) s += acc0[r] + acc1[r];
        lanePartial += s * (1.0f / 16.0f);       // columns duplicate rowsums 16x
    }
#else
    {
        float s = 0.0f;
        for (int d = 10; d < DIMS; ++d) {
            float df = SP(base + d) - SG(base + d);
            s += df * df;
        }
        lanePartial += s * m;
    }
#endif

    // ---- wave reduce (fixed order -> deterministic), one partial per tile ----
    float v = lanePartial;
    #pragma unroll
    for (int off = 16; off > 0; off >>= 1) v += __shfl_down(v, off, 32);
    if (lane == 0) part[tile] = v;

#undef SP
#undef SG
}

__global__ __launch_bounds__(256) void yolo_loss_reduce(
    const float* __restrict__ part, float* __restrict__ out)
{
    __shared__ float red[256];
    float s = 0.0f;
    for (int i = threadIdx.x; i < NTILES; i += 256) s += part[i];
    red[threadIdx.x] = s;
    __syncthreads();
    for (int o = 128; o > 0; o >>= 1) {
        if (threadIdx.x < o) red[threadIdx.x] += red[threadIdx.x + o];
        __syncthreads();
    }
    if (threadIdx.x == 0) out[0] = red[0] * (1.0f / 1024.0f);  // exact /B scale
}

extern "C" void kernel_launch(void* const* d_in, const int* in_sizes, int n_in,
                              void* d_out, int out_size, void* d_ws, size_t ws_size,
                              hipStream_t stream) {
    const float* p = (const float*)d_in[0];   // p_yolos, f32 [1024,14,14,90]
    const float* g = (const float*)d_in[1];   // g_yolos, f32 [1024,14,14,90]
    float* partials = (float*)d_ws;           // NTILES floats of scratch
    (void)in_sizes; (void)n_in; (void)out_size; (void)ws_size;

    yolo_loss_partial<<<NTILES / WPB, 32 * WPB, 0, stream>>>(p, g, partials);
    yolo_loss_reduce<<<1, 256, 0, stream>>>(partials, (float*)d_out);
}